// dctLayer_42511586296060
// MI455X (gfx1250) — compile-verified
//
#include <hip/hip_runtime.h>

typedef __attribute__((ext_vector_type(2))) float v2f;
typedef __attribute__((ext_vector_type(8))) float v8f;

#define IMG_W   1024
#define N_TILES (48 * 64 * 64)   // (16 b * 3 c) images, 64x64 tiles of 16x16

__device__ __forceinline__ float softround_px(float v) {
    // soft_round(v*255) - 128 ; jnp.round == round-half-even == rint
    float y = v * 255.0f;
    float r = __builtin_rintf(y);
    float d = y - r;
    return __builtin_fmaf(d * d, d, r) - 128.0f;
}

__global__ __launch_bounds__(256, 1) void dct8x8_wmma(
    const float* __restrict__ in, const float* __restrict__ dct,
    const float* __restrict__ idct, float* __restrict__ out)
{
    const int lane = threadIdx.x & 31;
    const int r16  = lane & 15;       // row for A-operand / N for B,C,D layouts
    const int hi   = lane >> 4;       // half-wave select
    const bool lo  = (hi == 0);

    // Per-lane constant operands (lane-dependent only; loaded once per wave).
    // Stage-1 B operand: Bblk = diag(iDCT,iDCT); element (K = 4q+v+2*hi, N = r16)
    // Stage-2 A operand: Ablk = diag(DCT,DCT);   element (M = r16, K = 4q+v+2*hi)
    v2f bconst[4], aconst[4];
#pragma unroll
    for (int q = 0; q < 4; ++q) {
#pragma unroll
        for (int v = 0; v < 2; ++v) {
            const int k = 4 * q + v + 2 * hi;
            const bool same = ((k < 8) == (r16 < 8));   // block-diagonal mask
            bconst[q][v] = same ? idct[(k & 7) * 8 + (r16 & 7)] : 0.0f;
            aconst[q][v] = same ? dct [(r16 & 7) * 8 + (k  & 7)] : 0.0f;
        }
    }

    const int wavesPerBlk = blockDim.x >> 5;
    const int nWaves      = (gridDim.x * blockDim.x) >> 5;
    const int wave        = blockIdx.x * wavesPerBlk + (threadIdx.x >> 5);

    for (int t = wave; t < N_TILES; t += nWaves) {
        const int img = t >> 12;          // tile / 4096
        const int rem = t & 4095;
        const int ty  = rem >> 6;
        const int tx  = rem & 63;
        const size_t base = (size_t)img * ((size_t)IMG_W * IMG_W)
                          + (size_t)(ty * 16) * IMG_W + (size_t)(tx * 16);

        // ---- Load X tile directly in A-operand layout (b64 pairs per lane) ----
        const float* src = in + base + (size_t)r16 * IMG_W + 2 * hi;
        v2f a[4];
#pragma unroll
        for (int q = 0; q < 4; ++q) {
            v2f x = __builtin_nontemporal_load((const v2f*)(src + 4 * q));
            x[0] = softround_px(x[0]);
            x[1] = softround_px(x[1]);
            a[q] = x;
        }

        // ---- Stage 1: T = X * diag(iDCT,iDCT) : accumulate 4x K=4 slices ----
        v8f tacc = {};
#pragma unroll
        for (int q = 0; q < 4; ++q)
            tacc = __builtin_amdgcn_wmma_f32_16x16x4_f32(
                false, a[q], false, bconst[q], (short)0, tacc, false, false);

        // ---- D-layout -> B-operand layout: half-wave swaps + selects ----
        float s0 = __shfl_xor(tacc[0], 16, 32);
        float s1 = __shfl_xor(tacc[1], 16, 32);
        float s2 = __shfl_xor(tacc[2], 16, 32);
        float s3 = __shfl_xor(tacc[3], 16, 32);
        float s4 = __shfl_xor(tacc[4], 16, 32);
        float s5 = __shfl_xor(tacc[5], 16, 32);
        float s6 = __shfl_xor(tacc[6], 16, 32);
        float s7 = __shfl_xor(tacc[7], 16, 32);

        v2f bt[4];
        bt[0][0] = lo ? tacc[0] : s2;   bt[0][1] = lo ? tacc[1] : s3;
        bt[1][0] = lo ? tacc[4] : s6;   bt[1][1] = lo ? tacc[5] : s7;
        bt[2][0] = lo ? s0 : tacc[2];   bt[2][1] = lo ? s1 : tacc[3];
        bt[3][0] = lo ? s4 : tacc[6];   bt[3][1] = lo ? s5 : tacc[7];

        // ---- Stage 2: Y = diag(DCT,DCT) * T ----
        v8f y = {};
#pragma unroll
        for (int q = 0; q < 4; ++q)
            y = __builtin_amdgcn_wmma_f32_16x16x4_f32(
                false, aconst[q], false, bt[q], (short)0, y, false, false);

        // ---- Store Y straight from C/D layout ----
        float* dst = out + base + (size_t)(hi * 8) * IMG_W + r16;
#pragma unroll
        for (int r = 0; r < 8; ++r)
            __builtin_nontemporal_store(y[r], dst + (size_t)r * IMG_W);
    }
}

extern "C" void kernel_launch(void* const* d_in, const int* in_sizes, int n_in,
                              void* d_out, int out_size, void* d_ws, size_t ws_size,
                              hipStream_t stream)
{
    const float* in   = (const float*)d_in[0];   // (16,3,1024,1024) fp32
    const float* DCT  = (const float*)d_in[1];   // 8x8 fp32
    const float* iDCT = (const float*)d_in[2];   // 8x8 fp32 (= DCT^T)
    float* out        = (float*)d_out;

    // 4096 blocks * 8 waves = 32768 waves; 196608 tiles -> exactly 6 per wave.
    dim3 grid(4096), block(256);
    hipLaunchKernelGGL(dct8x8_wmma, grid, block, 0, stream, in, DCT, iDCT, out);
}